// GraphSAGE_81638738362645
// MI455X (gfx1250) — compile-verified
//
#include <hip/hip_runtime.h>

// GraphSAGE fused kernel for gfx1250 (MI455X), fp32 WMMA (v_wmma_f32_16x16x4_f32).
//
// Block = 256 threads (8 wave32) owns 16 nodes.
//   Stage 1: wave w -> group (w&3), nodes (w>>2)*8 .. +7. Per node: gather
//            B = xs[group rows, idx[n,k]] (64x16), A = W1[g] (64x64 in regs),
//            64 WMMAs -> 64x16 acc; max over k (shfl_xor within 16-lane halves,
//            legal since bias+BN+ReLU are monotone), affine+relu -> LDS sMax.
//   Stage 2: 16 output tiles of 16 chan x 16 nodes, K=128. Tiles 0..7 read B
//            from global xs (concat rows < 256); tiles 8..15 read B from LDS
//            sMax (concat rows >= 256). 32 WMMAs per tile, affine+relu, store.

namespace {
constexpr int Cc = 256;
constexpr int Nn = 20000;
constexpr int Kk = 16;
constexpr int Cg = 64;
constexpr float kEps = 1e-5f;

typedef float v2f __attribute__((ext_vector_type(2)));
typedef float v8f __attribute__((ext_vector_type(8)));

__device__ __forceinline__ v8f wmma4(v2f a, v2f b, v8f c) {
  // D = A(16x4) * B(4x16) + C, fp32
  return __builtin_amdgcn_wmma_f32_16x16x4_f32(false, a, false, b, (short)0, c,
                                               false, false);
}
}  // namespace

__global__ __launch_bounds__(256) void sage_fused(
    const float* __restrict__ xs,        // (C, N)
    const long long* __restrict__ eidx,  // (2, N, K); use eidx[0] = first N*K
    const float* __restrict__ W1,        // (G, Cg, Cg)
    const float* __restrict__ b1, const float* __restrict__ g1,
    const float* __restrict__ be1, const float* __restrict__ m1,
    const float* __restrict__ v1,
    const float* __restrict__ W2,        // (G, Cg, 2*Cg)
    const float* __restrict__ b2, const float* __restrict__ g2,
    const float* __restrict__ be2, const float* __restrict__ m2,
    const float* __restrict__ v2,
    float* __restrict__ out)             // (C, N)
{
  __shared__ float sMax[Cc * 16];  // xj_max tile, [channel][node_local]
  __shared__ float sSc1[Cc], sSh1[Cc], sSc2[Cc], sSh2[Cc];

  const int tid = threadIdx.x;

  // Per-channel affine constants (bias folded through BN):
  //   relu((acc + b)*sc + (be - m*sc)) = relu(acc*sc + (b*sc + be - m*sc))
  {
    float sc = g1[tid] / sqrtf(v1[tid] + kEps);
    sSc1[tid] = sc;
    sSh1[tid] = b1[tid] * sc + be1[tid] - m1[tid] * sc;
    float sd = g2[tid] / sqrtf(v2[tid] + kEps);
    sSc2[tid] = sd;
    sSh2[tid] = b2[tid] * sd + be2[tid] - m2[tid] * sd;
  }
  __syncthreads();

  const int wave = tid >> 5;   // 0..7
  const int lane = tid & 31;
  const int m16  = lane & 15;  // M-row (A) / N-col (B,C,D) within tile
  const int h    = lane >> 4;  // lane half: selects K-pair / M+8 rows
  const int n0   = blockIdx.x * 16;

  // ---------------- Stage 1 ----------------
  const int g = wave & 3;
  const int gbase = g * Cg;

  // A fragments: W1[g] as 4 M-tiles x 16 K-chunks, float2 per lane per chunk.
  v2f aF[4][16];
  {
    const float* Wg = W1 + g * Cg * Cg;
#pragma unroll
    for (int t = 0; t < 4; ++t)
#pragma unroll
      for (int c = 0; c < 16; ++c)
        aF[t][c] = *(const v2f*)(Wg + (16 * t + m16) * Cg + 4 * c + 2 * h);
  }

  for (int nj = 0; nj < 8; ++nj) {
    const int nl = (wave >> 2) * 8 + nj;  // node_local 0..15
    const int n = n0 + nl;
    // neighbor column for this lane (col = m16); halves load duplicates
    const int j = (int)eidx[n * Kk + m16];
    const float* xcol = xs + j;

    v8f acc[4] = {v8f{}, v8f{}, v8f{}, v8f{}};
#pragma unroll
    for (int c = 0; c < 16; ++c) {
      const int row = gbase + 4 * c + 2 * h;
      v2f b;
      b.x = xcol[row * Nn];
      b.y = xcol[(row + 1) * Nn];
#pragma unroll
      for (int t = 0; t < 4; ++t) acc[t] = wmma4(aF[t][c], b, acc[t]);
    }

    // max over the 16 neighbor columns (lanes of each half), affine+relu
#pragma unroll
    for (int t = 0; t < 4; ++t) {
#pragma unroll
      for (int v = 0; v < 8; ++v) {
        float val = acc[t][v];
        val = fmaxf(val, __shfl_xor(val, 1));
        val = fmaxf(val, __shfl_xor(val, 2));
        val = fmaxf(val, __shfl_xor(val, 4));
        val = fmaxf(val, __shfl_xor(val, 8));
        if (m16 == v) {
          const int ch = gbase + 16 * t + v + 8 * h;
          sMax[ch * 16 + nl] = fmaxf(val * sSc1[ch] + sSh1[ch], 0.f);
        }
      }
    }
  }
  __syncthreads();

  // ---------------- Stage 2 ----------------
  // Tile mt covers channels 16*mt..16*mt+15; group = mt>>2; K = 128.
  // cat rows g*128..g*128+127: g in {0,1} -> xs rows (global);
  //                            g in {2,3} -> sMax rows (LDS).
  {
    // Tile mt = wave (groups 0..1): B from global xs, coalesced columns.
    const int mt = wave;
    const int gq = mt >> 2;
    const float* Wg2 = W2 + gq * Cg * 2 * Cg + ((mt & 3) * 16 + m16) * 2 * Cg;
    const float* xb = xs + (gq * 128) * Nn + n0 + m16;
    v8f acc = {};
#pragma unroll
    for (int c = 0; c < 32; ++c) {
      v2f a = *(const v2f*)(Wg2 + 4 * c + 2 * h);
      v2f b;
      b.x = xb[(4 * c + 2 * h) * Nn];
      b.y = xb[(4 * c + 2 * h + 1) * Nn];
      acc = wmma4(a, b, acc);
    }
#pragma unroll
    for (int v = 0; v < 8; ++v) {
      const int ch = 16 * mt + v + 8 * h;
      out[ch * Nn + n0 + m16] = fmaxf(acc[v] * sSc2[ch] + sSh2[ch], 0.f);
    }
  }
  {
    // Tile mt = wave + 8 (groups 2..3): B from LDS sMax.
    const int mt = wave + 8;
    const int gq = mt >> 2;
    const float* Wg2 = W2 + gq * Cg * 2 * Cg + ((mt & 3) * 16 + m16) * 2 * Cg;
    const float* sb = sMax + ((gq - 2) * 128) * 16 + m16;
    v8f acc = {};
#pragma unroll
    for (int c = 0; c < 32; ++c) {
      v2f a = *(const v2f*)(Wg2 + 4 * c + 2 * h);
      v2f b;
      b.x = sb[(4 * c + 2 * h) * 16];
      b.y = sb[(4 * c + 2 * h + 1) * 16];
      acc = wmma4(a, b, acc);
    }
#pragma unroll
    for (int v = 0; v < 8; ++v) {
      const int ch = 16 * mt + v + 8 * h;
      out[ch * Nn + n0 + m16] = fmaxf(acc[v] * sSc2[ch] + sSh2[ch], 0.f);
    }
  }
}

extern "C" void kernel_launch(void* const* d_in, const int* in_sizes, int n_in,
                              void* d_out, int out_size, void* d_ws,
                              size_t ws_size, hipStream_t stream) {
  const float* xs = (const float*)d_in[0];
  const long long* eidx = (const long long*)d_in[1];
  const float* W1 = (const float*)d_in[2];
  const float* b1 = (const float*)d_in[3];
  const float* g1 = (const float*)d_in[4];
  const float* be1 = (const float*)d_in[5];
  const float* m1 = (const float*)d_in[6];
  const float* v1 = (const float*)d_in[7];
  const float* W2 = (const float*)d_in[8];
  const float* b2 = (const float*)d_in[9];
  const float* g2 = (const float*)d_in[10];
  const float* be2 = (const float*)d_in[11];
  const float* m2 = (const float*)d_in[12];
  const float* v2 = (const float*)d_in[13];

  dim3 grid(Nn / 16);  // 1250 blocks x 16 nodes
  sage_fused<<<grid, 256, 0, stream>>>(xs, eidx, W1, b1, g1, be1, m1, v1, W2,
                                       b2, g2, be2, m2, v2, (float*)d_out);
}